// MS_SSIM_78692390797609
// MI455X (gfx1250) — compile-verified
//
#include <hip/hip_runtime.h>
#include <cmath>

// ---------------------------------------------------------------------------
// MS-SSIM on gfx1250 (MI455X), fp32 path via V_WMMA_F32_16X16X4_F32.
// One wave32 per 16x16 output tile; separable 11-tap Gaussian conv as banded
// WMMA products. Patch is zero-padded in LDS (28x32) so all inner fragment
// loads are unguarded base+immediate ds_load_b32; weight fragments built once
// and shared by both passes; all five signals accumulated in one sweep;
// fused SSIM epilogue + reduction; avg-pool pyramid; finalize kernel.
// ---------------------------------------------------------------------------

typedef __attribute__((ext_vector_type(2))) float v2f;
typedef __attribute__((ext_vector_type(8))) float v8f;

struct GaussW { float g[11]; };

__device__ __forceinline__ v8f wmma_f32_16x16x4(v2f a, v2f b, v8f c) {
    // 8 args: (neg_a, A, neg_b, B, c_mod, C, reuse_a, reuse_b)
    return __builtin_amdgcn_wmma_f32_16x16x4_f32(
        false, a, false, b, (short)0, c, false, false);
}

#define SSIM_C1 0.0001f   // (0.01*1)^2
#define SSIM_C2 0.0009f   // (0.03*1)^2

__global__ __launch_bounds__(32)
void ssim_scale_kernel(const float* __restrict__ X, const float* __restrict__ Y,
                       int H, float* __restrict__ accum, int scale, GaussW gw)
{
    const int W    = H;        // square images
    const int outD = H - 10;   // valid conv output dim
    const int tid    = threadIdx.x;
    const int lane16 = tid & 15;
    const int hh     = tid >> 4;       // lane half (0 or 1)
    const int r0 = blockIdx.y * 16;
    const int c0 = blockIdx.x * 16;
    const int plane = blockIdx.z;      // b*3 + c, 0..47

    __shared__ float gLds[16];         // 11 taps, zero-padded
    __shared__ float pX[28 * 32];      // patch x, 28 rows x 32 cols, zero-pad
    __shared__ float pY[28 * 32];      // patch y
    __shared__ float Vlds[5][16 * 33]; // per-signal vertical results

    if (tid < 16) gLds[tid] = (tid < 11) ? gw.g[tid] : 0.0f;

    const float* xp = X + (size_t)plane * H * W;
    const float* yp = Y + (size_t)plane * H * W;

    // ---- load 26x26 patch into zero-padded 28x32 LDS tile -----------------
    // 896/32 = 28 uniform iterations per lane, no divergence at loop level.
    for (int idx = tid; idx < 28 * 32; idx += 32) {
        int i = idx >> 5, j = idx & 31;
        int gr = r0 + i, gc = c0 + j;
        bool ok = (i < 26) && (j < 26) && (gr < H) && (gc < W);
        size_t go = ok ? ((size_t)gr * W + gc) : 0;
        float vx = xp[go]; float vy = yp[go];
        pX[idx] = ok ? vx : 0.0f;
        pY[idx] = ok ? vy : 0.0f;
    }
    __syncthreads();

    // ---- banded Gaussian fragments: band(k - lane16), k = 4t + 2*hh + j ---
    // Identical layout serves as vertical-pass A and horizontal-pass B.
    v2f wFrag[7];
    #pragma unroll
    for (int t = 0; t < 7; ++t) {
        #pragma unroll
        for (int j = 0; j < 2; ++j) {
            int k = 4 * t + 2 * hh + j;              // 0..27
            int d = k - lane16;                      // band offset
            int idx = ((unsigned)d < 11u) ? d : 11;  // 11 -> padded zero
            wFrag[t][j] = gLds[idx];
        }
    }

    // ---- vertical pass: all five signals in one sweep over the patch ------
    // Fragment element address: pbase + t*128 + j*32 + cb*16 (imm offsets).
    const int pbase = hh * 64 + lane16;
    v8f accV[5][2];
    #pragma unroll
    for (int s = 0; s < 5; ++s)
        #pragma unroll
        for (int cb = 0; cb < 2; ++cb)
            accV[s][cb] = (v8f){0.f,0.f,0.f,0.f,0.f,0.f,0.f,0.f};

    #pragma unroll
    for (int t = 0; t < 7; ++t) {
        v2f a = wFrag[t];
        #pragma unroll
        for (int cb = 0; cb < 2; ++cb) {
            v2f b0, b1, b2, b3, b4;
            #pragma unroll
            for (int j = 0; j < 2; ++j) {
                int li = pbase + t * 128 + j * 32 + cb * 16;  // always in-bounds
                float px = pX[li], py = pY[li];               // zeros in pad
                b0[j] = px; b1[j] = py;
                b2[j] = px * px; b3[j] = py * py; b4[j] = px * py;
            }
            accV[0][cb] = wmma_f32_16x16x4(a, b0, accV[0][cb]);
            accV[1][cb] = wmma_f32_16x16x4(a, b1, accV[1][cb]);
            accV[2][cb] = wmma_f32_16x16x4(a, b2, accV[2][cb]);
            accV[3][cb] = wmma_f32_16x16x4(a, b3, accV[3][cb]);
            accV[4][cb] = wmma_f32_16x16x4(a, b4, accV[4][cb]);
        }
    }

    // ---- C/D layout -> A layout via LDS -----------------------------------
    // Store address: (8*hh*33 + lane16) + i*33 + cb*16 + s*528 (imm offsets).
    const int vsbase = 8 * hh * 33 + lane16;
    #pragma unroll
    for (int s = 0; s < 5; ++s) {
        #pragma unroll
        for (int i = 0; i < 8; ++i) {
            Vlds[0][vsbase + s * 528 + i * 33]      = accV[s][0][i];
            Vlds[0][vsbase + s * 528 + i * 33 + 16] = accV[s][1][i]; // cols 26..31 = 0
        }
    }
    __syncthreads();

    // ---- horizontal pass ---------------------------------------------------
    // Load address: (lane16*33 + 2*hh) + s*528 + t*4 + j (imm offsets).
    const int vlbase = lane16 * 33 + 2 * hh;
    v8f accO[5];
    #pragma unroll
    for (int s = 0; s < 5; ++s) {
        v8f acc = (v8f){0.f,0.f,0.f,0.f,0.f,0.f,0.f,0.f};
        #pragma unroll
        for (int t = 0; t < 7; ++t) {
            v2f a;
            #pragma unroll
            for (int j = 0; j < 2; ++j) {
                a[j] = Vlds[0][vlbase + s * 528 + t * 4 + j]; // zeros past col 25
            }
            acc = wmma_f32_16x16x4(a, wFrag[t], acc);
        }
        accO[s] = acc;
    }

    // ---- SSIM epilogue + masked reduction ---------------------------------
    float csSum = 0.0f, ssSum = 0.0f;
    int ocol = c0 + lane16;
    #pragma unroll
    for (int i = 0; i < 8; ++i) {
        int orow = r0 + i + 8 * hh;
        bool ok = (orow < outD) && (ocol < outD);
        float mu1 = accO[0][i], mu2 = accO[1][i];
        float exx = accO[2][i], eyy = accO[3][i], exy = accO[4][i];
        float s1  = exx - mu1 * mu1;
        float s2  = eyy - mu2 * mu2;
        float s12 = exy - mu1 * mu2;
        float den = s1 + s2 + SSIM_C2;
        float num = 2.0f * s12 + SSIM_C2;
        float cs  = num / den;
        float ss  = (2.0f * mu1 * mu2 + SSIM_C1) * num /
                    ((mu1 * mu1 + mu2 * mu2 + SSIM_C1) * den);
        csSum += ok ? cs : 0.0f;
        ssSum += ok ? ss : 0.0f;
    }
    #pragma unroll
    for (int o = 16; o > 0; o >>= 1) {
        csSum += __shfl_down(csSum, o, 32);
        ssSum += __shfl_down(ssSum, o, 32);
    }
    if (tid == 0) {
        atomicAdd(&accum[2 * scale + 0], csSum);
        atomicAdd(&accum[2 * scale + 1], ssSum);
    }
}

// 2x2 average pool, stride 2, floor mode; both signals in one pass.
__global__ void avgpool2_kernel(const float* __restrict__ inX,
                                const float* __restrict__ inY,
                                float* __restrict__ outX,
                                float* __restrict__ outY,
                                int H, int OH)
{
    int n = 48 * OH * OH;
    int idx = blockIdx.x * blockDim.x + threadIdx.x;
    if (idx >= n) return;
    int p   = idx / (OH * OH);
    int rem = idx - p * OH * OH;
    int r = rem / OH, c = rem - r * OH;
    const float* xi = inX + (size_t)p * H * H;
    const float* yi = inY + (size_t)p * H * H;
    int i2 = 2 * r, j2 = 2 * c;
    float sx = xi[i2*H+j2] + xi[i2*H+j2+1] + xi[(i2+1)*H+j2] + xi[(i2+1)*H+j2+1];
    float sy = yi[i2*H+j2] + yi[i2*H+j2+1] + yi[(i2+1)*H+j2] + yi[(i2+1)*H+j2+1];
    outX[idx] = 0.25f * sx;
    outY[idx] = 0.25f * sy;
}

// mean -> relu -> (x+1)/2 -> weighted product -> 1 - ms_ssim
__global__ void finalize_kernel(const float* __restrict__ accum,
                                float* __restrict__ out)
{
    const float w[5]  = {0.0448f, 0.2856f, 0.3001f, 0.2363f, 0.1333f};
    const int   od[5] = {502, 246, 118, 54, 22};
    float ms = 1.0f;
    for (int i = 0; i < 5; ++i) {
        float cnt = 48.0f * (float)od[i] * (float)od[i];
        float cs = fmaxf(accum[2*i+0] / cnt, 0.0f);
        float ss = fmaxf(accum[2*i+1] / cnt, 0.0f);
        float csn = 0.5f * (cs + 1.0f);
        float ssn = 0.5f * (ss + 1.0f);
        ms *= powf((i < 4) ? csn : ssn, w[i]);
    }
    out[0] = 1.0f - ms;
}

extern "C" void kernel_launch(void* const* d_in, const int* in_sizes, int n_in,
                              void* d_out, int out_size, void* d_ws, size_t ws_size,
                              hipStream_t stream)
{
    const float* X0 = (const float*)d_in[0];
    const float* Y0 = (const float*)d_in[1];
    float* ws    = (float*)d_ws;
    float* accum = ws;                       // 10 floats used, 64 reserved

    hipMemsetAsync(accum, 0, 64 * sizeof(float), stream);

    // Gaussian window, sigma=1.5, size 11, normalized (host-computed, determ.)
    GaussW gw;
    {
        double g[11], sum = 0.0;
        for (int i = 0; i < 11; ++i) { double d = (double)(i - 5); g[i] = exp(-d * d / 4.5); sum += g[i]; }
        for (int i = 0; i < 11; ++i) gw.g[i] = (float)(g[i] / sum);
    }

    static const int IN[5]  = {512, 256, 128, 64, 32};
    static const int OUT[5] = {502, 246, 118, 54, 22};

    // pyramid buffers in workspace (floats), after 64-float accumulator area
    float* bufX[5] = {nullptr, nullptr, nullptr, nullptr, nullptr};
    float* bufY[5] = {nullptr, nullptr, nullptr, nullptr, nullptr};
    size_t off = 64;
    for (int s = 1; s < 5; ++s) {
        size_t n = (size_t)48 * IN[s] * IN[s];
        bufX[s] = ws + off; off += n;
        bufY[s] = ws + off; off += n;
    }

    for (int s = 0; s < 5; ++s) {
        const float* X = (s == 0) ? X0 : bufX[s];
        const float* Y = (s == 0) ? Y0 : bufY[s];
        int H = IN[s], O = OUT[s];
        dim3 grid((O + 15) / 16, (O + 15) / 16, 48);
        ssim_scale_kernel<<<grid, 32, 0, stream>>>(X, Y, H, accum, s, gw);
        if (s < 4) {
            int OH = IN[s + 1];
            int n  = 48 * OH * OH;
            avgpool2_kernel<<<(n + 255) / 256, 256, 0, stream>>>(
                X, Y, bufX[s + 1], bufY[s + 1], H, OH);
        }
    }

    finalize_kernel<<<1, 1, 0, stream>>>(accum, (float*)d_out);
}